// Dynamic2Select_13932873908843
// MI455X (gfx1250) — compile-verified
//
#include <hip/hip_runtime.h>
#include <stdint.h>

// Problem constants from the reference.
#define Bc 2
#define Nc 5
#define Cc 64
#define Ac 2
#define Hc 128
#define Wc 256
#define HWc (Hc * Wc)
#define CSPLIT 4
#define CCHUNK (Cc / CSPLIT)

__device__ __forceinline__ int iclampi(int v, int lo, int hi) {
    return v < lo ? lo : (v > hi ? hi : v);
}

// ---------------------------------------------------------------------------
// Kernel 1: owner selection (top-2 nearest agents per batch) + zero counters.
// ws layout (floats): [0..9] owner, [10..19] num, [20..29] den
// ---------------------------------------------------------------------------
__global__ __launch_bounds__(64) void prep_kernel(const float* __restrict__ nam,
                                                  float* __restrict__ ws) {
    const int t = threadIdx.x;
    if (t < Bc * Nc) { ws[10 + t] = 0.0f; ws[20 + t] = 0.0f; }
    if (t < Bc) {
        const int b = t;
        float d[Nc];
        for (int n = 0; n < Nc; ++n) {
            const float* m = nam + ((size_t)(b * Nc + 0) * Nc + n) * 6;
            const float tx = m[2], ty = m[5];
            d[n] = tx * tx + ty * ty;
        }
        d[0] = __builtin_inff();
        // top-2 smallest, stable ties (lower index wins) like lax.top_k(-d)
        int i1 = 0; float b1 = __builtin_inff();
        for (int n = 0; n < Nc; ++n) if (d[n] < b1) { b1 = d[n]; i1 = n; }
        int i2 = 0; float b2 = __builtin_inff();
        for (int n = 0; n < Nc; ++n) if (n != i1 && d[n] < b2) { b2 = d[n]; i2 = n; }
        for (int n = 0; n < Nc; ++n)
            ws[b * Nc + n] = (n == 0 || n == i1 || n == i2) ? 1.0f : 0.0f;
    }
}

// ---------------------------------------------------------------------------
// Kernel 2: fused warp + comm-mask + owner-gate + channel-max + vis counts.
// grid = (H, B, CSPLIT); block = W (256 threads = 8 wave32 = one output row).
// ---------------------------------------------------------------------------
__global__ __launch_bounds__(Wc) void warp_fuse_kernel(
    const float* __restrict__ x, const float* __restrict__ psm,
    const float* __restrict__ nam, const float* __restrict__ ws,
    float* __restrict__ out, float* __restrict__ num, float* __restrict__ den) {
    const int h  = blockIdx.x;
    const int b  = blockIdx.y;
    const int cz = blockIdx.z;
    const int w  = threadIdx.x;

    __shared__ __align__(16) float s_theta[32];  // 30 used: nam[b,0,:,:,:]
    __shared__ __align__(16) float s_owner[8];   // 5 used

    // Never-taken, runtime-guarded stores: the compiler cannot prove nam!=NULL,
    // so it must keep these stores alive. This prevents "shared array has no
    // stores -> loads are undef" folding while keeping the accesses in
    // addrspace(3) (real ds_load/ds_store), since the real producer below is
    // an async DMA the compiler cannot see.
    if ((uintptr_t)nam == 0) {
        s_theta[w & 31] = 0.0f;
        s_owner[w & 7]  = 0.0f;
    }

    // --- CDNA5 async global->LDS staging of theta + owner (ASYNCcnt path) ---
    {
        const int t = threadIdx.x;
        if (t < 15) {  // 15 x b64 = 30 floats of theta
            uint32_t lo = (uint32_t)(uintptr_t)(&s_theta[t * 2]);
            const void* g = (const void*)(nam + (size_t)b * Nc * Nc * 6 + (size_t)t * 2);
            asm volatile("global_load_async_to_lds_b64 %0, %1, off"
                         :: "v"(lo), "v"(g) : "memory");
        } else if (t >= 16 && t < 16 + Nc) {  // 5 x b32 owner flags
            const int n = t - 16;
            uint32_t lo = (uint32_t)(uintptr_t)(&s_owner[n]);
            const void* g = (const void*)(ws + b * Nc + n);
            asm volatile("global_load_async_to_lds_b32 %0, %1, off"
                         :: "v"(lo), "v"(g) : "memory");
        }
        asm volatile("s_wait_asynccnt 0x0" ::: "memory");
        __syncthreads();
    }

    // align_corners=True grid (matches jnp.linspace(-1,1,.))
    const float gx = -1.0f + (float)w * (2.0f / 255.0f);
    const float gy = -1.0f + (float)h * (2.0f / 127.0f);

    int   o00[Nc], o01[Nc], o10[Nc], o11[Nc];
    float q00[Nc], q01[Nc], q10[Nc], q11[Nc];
    bool  act[Nc];
    bool  anyInactive = false;
    const int lane = threadIdx.x & 31;

#pragma unroll
    for (int n = 0; n < Nc; ++n) {
        const float t00 = s_theta[n * 6 + 0], t01 = s_theta[n * 6 + 1], t02 = s_theta[n * 6 + 2];
        const float t10 = s_theta[n * 6 + 3], t11 = s_theta[n * 6 + 4], t12 = s_theta[n * 6 + 5];
        const float cx = t00 * gx + t01 * gy + t02;
        const float cy = t10 * gx + t11 * gy + t12;
        const float ix = ((cx + 1.0f) * 0.5f) * 255.0f;
        const float iy = ((cy + 1.0f) * 0.5f) * 127.0f;

        // nearest-mode visibility (jnp.round = half-to-even = rintf)
        const int xn = (int)rintf(ix);
        const int yn = (int)rintf(iy);
        const bool vis = (yn >= 0) && (yn < Hc) && (xn >= 0) && (xn < Wc);

        // bilinear taps with zero-padding folded into weights
        const float xf = floorf(ix), yf = floorf(iy);
        const float fx = ix - xf,    fy = iy - yf;
        const int x0 = (int)xf, y0 = (int)yf, x1 = x0 + 1, y1 = y0 + 1;
        const bool bx0 = (x0 >= 0) && (x0 < Wc), bx1 = (x1 >= 0) && (x1 < Wc);
        const bool by0 = (y0 >= 0) && (y0 < Hc), by1 = (y1 >= 0) && (y1 < Hc);
        const int cx0 = iclampi(x0, 0, Wc - 1), cx1 = iclampi(x1, 0, Wc - 1);
        const int cy0 = iclampi(y0, 0, Hc - 1), cy1 = iclampi(y1, 0, Hc - 1);
        float w00 = (1.0f - fy) * (1.0f - fx); if (!(by0 && bx0)) w00 = 0.0f;
        float w01 = (1.0f - fy) * fx;          if (!(by0 && bx1)) w01 = 0.0f;
        float w10 = fy * (1.0f - fx);          if (!(by1 && bx0)) w10 = 0.0f;
        float w11 = fy * fx;                   if (!(by1 && bx1)) w11 = 0.0f;
        o00[n] = cy0 * Wc + cx0; o01[n] = cy0 * Wc + cx1;
        o10[n] = cy1 * Wc + cx0; o11[n] = cy1 * Wc + cx1;
        q00[n] = w00; q01[n] = w01; q10[n] = w10; q11[n] = w11;

        // confidence -> comm mask: max over A of sigmoid == sigmoid(max)
        const float* pp = psm + (size_t)(b * Nc + n) * Ac * HWc;
        const float c0 = w00 * pp[o00[n]] + w01 * pp[o01[n]] +
                         w10 * pp[o10[n]] + w11 * pp[o11[n]];
        const float c1 = w00 * pp[HWc + o00[n]] + w01 * pp[HWc + o01[n]] +
                         w10 * pp[HWc + o10[n]] + w11 * pp[HWc + o11[n]];
        const float m  = fmaxf(c0, c1);
        const float sg = 1.0f / (1.0f + expf(-m));
        const bool comm = (n == 0) || (sg > 0.01f);
        const bool a = comm && (s_owner[n] > 0.5f);
        act[n] = a;
        anyInactive |= !a;

        // exact integer counts via wave32 ballot+popcount, once per pixel
        if (cz == 0) {
            const unsigned long long mv = __ballot((int)vis);
            const unsigned long long mf = __ballot((int)(a && vis));
            if (lane == 0) {
                atomicAdd(&den[b * Nc + n], (float)__popcll(mv));
                atomicAdd(&num[b * Nc + n], (float)__popcll(mf));
            }
        }
    }

    // channel loop: max over agents; inactive agents contribute exact 0
    const int cbeg = cz * CCHUNK;
#pragma unroll 4
    for (int c = cbeg; c < cbeg + CCHUNK; ++c) {
#pragma unroll
        for (int pn = 0; pn < Nc; ++pn) {  // stream-ahead prefetch (global_prefetch_b8)
            if (act[pn] && (c + 2 < cbeg + CCHUNK)) {
                const float* p = x + ((size_t)(b * Nc + pn) * Cc + c + 2) * HWc;
                __builtin_prefetch(p + o00[pn], 0, 1);
            }
        }
        float acc = -3.402823466e+38f;  // n=0 is always active
#pragma unroll
        for (int n = 0; n < Nc; ++n) {
            if (act[n]) {
                const float* p = x + ((size_t)(b * Nc + n) * Cc + c) * HWc;
                const float v = q00[n] * p[o00[n]] + q01[n] * p[o01[n]] +
                                q10[n] * p[o10[n]] + q11[n] * p[o11[n]];
                acc = fmaxf(acc, v);
            }
        }
        if (anyInactive) acc = fmaxf(acc, 0.0f);
        out[((size_t)(b * Cc + c) * Hc + h) * Wc + w] = acc;
    }
}

// ---------------------------------------------------------------------------
// Kernel 3: effective_rate = mean_j( num_j / max(den_j, 1) )
// ---------------------------------------------------------------------------
__global__ void rate_kernel(const float* __restrict__ ws, float* __restrict__ out_rate) {
    if (threadIdx.x == 0 && blockIdx.x == 0) {
        float s = 0.0f;
        for (int j = 0; j < Bc * Nc; ++j) {
            const float nm = ws[10 + j];
            const float dn = fmaxf(ws[20 + j], 1.0f);
            s += nm / dn;
        }
        *out_rate = s / (float)(Bc * Nc);
    }
}

extern "C" void kernel_launch(void* const* d_in, const int* in_sizes, int n_in,
                              void* d_out, int out_size, void* d_ws, size_t ws_size,
                              hipStream_t stream) {
    const float* x   = (const float*)d_in[0];  // (10,64,128,256)
    const float* psm = (const float*)d_in[1];  // (10,2,128,256)
    const float* nam = (const float*)d_in[2];  // (2,5,5,2,3)
    float* out = (float*)d_out;                // 2*64*128*256 + 1
    float* ws  = (float*)d_ws;                 // 30 floats used

    prep_kernel<<<1, 64, 0, stream>>>(nam, ws);
    dim3 grid(Hc, Bc, CSPLIT);
    warp_fuse_kernel<<<grid, Wc, 0, stream>>>(x, psm, nam, ws, out, ws + 10, ws + 20);
    rate_kernel<<<1, 32, 0, stream>>>(ws, out + (size_t)Bc * Cc * Hc * Wc);
}